// MHCCoeffs_25640954757693
// MI455X (gfx1250) — compile-verified
//
#include <hip/hip_runtime.h>
#include <math.h>

typedef float v2f __attribute__((ext_vector_type(2)));
typedef float v8f __attribute__((ext_vector_type(8)));

#define D_IN         8192
#define NTOK         8192            // B*T = 2*4096
#define NCOL         32              // 24 real cols (4 pre, 4 post, 16 res) + 8 pad
#define KC           128             // K chunk staged through LDS
#define XS           (KC + 4)        // 132 dwords; 132 % 64 == 4 -> conflict-free b64 frag reads
#define WAVES        4
#define THREADS     (WAVES * 32)
#define TOK_PER_WAVE 16
#define TOK_PER_WG   (WAVES * TOK_PER_WAVE)   // 64
#define NBLK         (NTOK / TOK_PER_WG)      // 128
#define NCH          (D_IN / KC)              // 64 chunks
#define BUF_FLOATS   ((TOK_PER_WG + NCOL) * XS)  // 12672 floats per buffer
#define SCS          36              // epilogue scratch row stride (32 dots + 2 sumsq + pad)
#define EPS          1e-6f

// ---------------------------------------------------------------------------
// Kernel 1: fold rms_w into all 24 weight rows, pad to 32 rows, write to d_ws.
// ---------------------------------------------------------------------------
__global__ void mhc_pack_weights(const float* __restrict__ rms_w,
                                 const float* __restrict__ w_pre,
                                 const float* __restrict__ w_post,
                                 const float* __restrict__ w_res,
                                 float* __restrict__ wc) {
  int idx = blockIdx.x * blockDim.x + threadIdx.x;   // 0 .. 32*8192
  int n = idx >> 13;
  int d = idx & (D_IN - 1);
  float w;
  if      (n < 4)  w = w_pre [(long)n * D_IN + d];
  else if (n < 8)  w = w_post[(long)(n - 4) * D_IN + d];
  else if (n < 24) w = w_res [(long)(n - 8) * D_IN + d];
  else             w = 0.0f;
  wc[idx] = w * rms_w[d];
}

// ---------------------------------------------------------------------------
// Kernel 2: fused rmsnorm + 24-column GEMM (fp32 WMMA 16x16x4) with
// double-buffered async global->LDS staging (ASYNCcnt), sigmoid/Sinkhorn
// epilogue. Each wave owns 16 tokens and accumulates over full K.
// ---------------------------------------------------------------------------
__global__ void __launch_bounds__(THREADS)
mhc_main(const float* __restrict__ x,
         const float* __restrict__ wc,
         const float* __restrict__ b_pre,
         const float* __restrict__ b_post,
         const float* __restrict__ b_res,
         const float* __restrict__ a_pre,
         const float* __restrict__ a_post,
         const float* __restrict__ a_res,
         float* __restrict__ out) {
  extern __shared__ float lds[];

  const int tid  = threadIdx.x;
  const int wave = tid >> 5;
  const int lane = tid & 31;
  const int n15  = lane & 15;           // M (A) / N (B,C) low index
  const int h    = lane >> 4;           // K-half selector for A/B fragments
  const int tok0 = blockIdx.x * TOK_PER_WG;
  const int arow = wave * TOK_PER_WAVE + n15;

  v8f acc0 = {0.f,0.f,0.f,0.f,0.f,0.f,0.f,0.f};
  v8f acc1 = {0.f,0.f,0.f,0.f,0.f,0.f,0.f,0.f};
  float sumsq = 0.0f;

  // Issue async global->LDS copies for chunk `ch` into buffer `buf`.
  // GVS mode: addr = SADDR(base) + VADDR(byte offset); LDS dest = raw byte off.
  auto issue_chunk = [&](int ch, int buf) {
    const unsigned lbase = (unsigned)(buf * BUF_FLOATS * 4);
    // X tile: 64 rows x 32 float4 (16 b128 per thread)
    for (int i = tid; i < TOK_PER_WG * (KC / 4); i += THREADS) {
      int row = i >> 5;
      int c4  = i & 31;
      unsigned go = (unsigned)(((unsigned)(tok0 + row) * D_IN + ch * KC + c4 * 4) * 4u);
      unsigned lo = lbase + (unsigned)((row * XS + c4 * 4) * 4);
      asm volatile("global_load_async_to_lds_b128 %0, %1, %2"
                   :: "v"(lo), "v"(go), "s"(x) : "memory");
    }
    // packed-weight tile: 32 rows x 32 float4 (8 b128 per thread)
    const unsigned lwb = lbase + (unsigned)(TOK_PER_WG * XS * 4);
    for (int i = tid; i < NCOL * (KC / 4); i += THREADS) {
      int row = i >> 5;
      int c4  = i & 31;
      unsigned go = (unsigned)(((unsigned)row * D_IN + ch * KC + c4 * 4) * 4u);
      unsigned lo = lwb + (unsigned)((row * XS + c4 * 4) * 4);
      asm volatile("global_load_async_to_lds_b128 %0, %1, %2"
                   :: "v"(lo), "v"(go), "s"(wc) : "memory");
    }
  };

  issue_chunk(0, 0);

  for (int ch = 0; ch < NCH; ++ch) {
    // Wait own async copies of buf[ch&1]; barrier makes everyone's visible.
    asm volatile("s_wait_asynccnt 0x0" ::: "memory");
    __syncthreads();
    // Prefetch next chunk into the other buffer (safe: that buffer was last
    // read in iteration ch-1, which every wave finished before this barrier).
    if (ch + 1 < NCH) issue_chunk(ch + 1, (ch + 1) & 1);

    const float* Xs = lds + (ch & 1) * BUF_FLOATS;
    const float* Bs = Xs + TOK_PER_WG * XS;

    // ---- fp32 WMMA over the chunk: 32 K-steps, 2 N-tiles each ----
#pragma unroll 8
    for (int k0 = 0; k0 < KC; k0 += 4) {
      v2f a  = *(const v2f*)&Xs[arow * XS + k0 + 2 * h];          // A(16x4)
      v2f b0 = *(const v2f*)&Bs[n15 * XS + k0 + 2 * h];           // B cols 0..15
      v2f b1 = *(const v2f*)&Bs[(16 + n15) * XS + k0 + 2 * h];    // B cols 16..31
      sumsq += a[0] * a[0] + a[1] * a[1];                          // fused rmsnorm stat
      acc0 = __builtin_amdgcn_wmma_f32_16x16x4_f32(false, a, false, b0,
                                                   (short)0, acc0, false, false);
      acc1 = __builtin_amdgcn_wmma_f32_16x16x4_f32(false, a, false, b1,
                                                   (short)0, acc1, false, false);
    }
  }

  __syncthreads();   // all compute done; reuse buffer 0 as epilogue scratch
  float* sc = lds;   // rows partitioned per wave: [wave*16 + m][SCS]
#pragma unroll
  for (int v = 0; v < 8; ++v) {
    int m = v + 8 * h;                              // C/D layout: lane = n + 16*(m>=8)
    sc[(wave * 16 + m) * SCS + n15]      = acc0[v];
    sc[(wave * 16 + m) * SCS + 16 + n15] = acc1[v];
  }
  sc[(wave * 16 + n15) * SCS + 32 + h] = sumsq;     // two K-halves per token

  // ---- per-token epilogue: 16 lanes/wave, one token each ----
  if (lane < 16) {
    const int tloc = wave * 16 + lane;
    const int tok  = tok0 + tloc;
    const float* dd = &sc[tloc * SCS];
    float ss = dd[32] + dd[33];
    float s  = rsqrtf(ss * (1.0f / (float)D_IN) + EPS);   // rsqrt(mean(x^2)+eps)
    float apre = a_pre[0], apost = a_post[0], ares = a_res[0];

#pragma unroll
    for (int j = 0; j < 4; ++j) {
      float z = apre * s * dd[j] + b_pre[j];
      out[(long)tok * 4 + j] = 1.0f / (1.0f + expf(-z));
    }
#pragma unroll
    for (int j = 0; j < 4; ++j) {
      float z = apost * s * dd[4 + j] + b_post[j];
      out[(long)NTOK * 4 + (long)tok * 4 + j] = 2.0f / (1.0f + expf(-z));
    }

    // Sinkhorn-Knopp on the 4x4 res block, fully in registers
    float M[16];
#pragma unroll
    for (int i = 0; i < 4; ++i) {
      float z[4];
      float mx = -3.402823466e38f;
#pragma unroll
      for (int j = 0; j < 4; ++j) {
        z[j] = ares * s * dd[8 + 4 * i + j] + b_res[4 * i + j];
        mx = fmaxf(mx, z[j]);
      }
#pragma unroll
      for (int j = 0; j < 4; ++j)
        M[4 * i + j] = fmaxf(expf(z[j] - mx), EPS);
    }
    for (int it = 0; it < 20; ++it) {
#pragma unroll
      for (int i = 0; i < 4; ++i) {
        float rs = M[4*i] + M[4*i+1] + M[4*i+2] + M[4*i+3];
        float r  = 1.0f / fmaxf(rs, EPS);
        M[4*i] *= r; M[4*i+1] *= r; M[4*i+2] *= r; M[4*i+3] *= r;
      }
#pragma unroll
      for (int j = 0; j < 4; ++j) {
        float cs = M[j] + M[4+j] + M[8+j] + M[12+j];
        float r  = 1.0f / fmaxf(cs, EPS);
        M[j] *= r; M[4+j] *= r; M[8+j] *= r; M[12+j] *= r;
      }
    }
    float* hres = out + (long)NTOK * 8 + (long)tok * 16;
#pragma unroll
    for (int q = 0; q < 16; ++q) hres[q] = M[q];
  }
}

// ---------------------------------------------------------------------------
extern "C" void kernel_launch(void* const* d_in, const int* in_sizes, int n_in,
                              void* d_out, int out_size, void* d_ws, size_t ws_size,
                              hipStream_t stream) {
  (void)in_sizes; (void)n_in; (void)out_size; (void)ws_size;
  const float* x      = (const float*)d_in[0];   // [2,4096,4,2048] = [8192][8192]
  const float* rms_w  = (const float*)d_in[1];
  const float* w_pre  = (const float*)d_in[2];
  const float* w_post = (const float*)d_in[3];
  const float* w_res  = (const float*)d_in[4];
  const float* b_pre  = (const float*)d_in[5];
  const float* b_post = (const float*)d_in[6];
  const float* b_res  = (const float*)d_in[7];
  const float* a_pre  = (const float*)d_in[8];
  const float* a_post = (const float*)d_in[9];
  const float* a_res  = (const float*)d_in[10];
  float* wcw = (float*)d_ws;                     // 32*8192 floats = 1 MB scratch
  float* out = (float*)d_out;

  hipLaunchKernelGGL(mhc_pack_weights, dim3((NCOL * D_IN) / 256), dim3(256), 0, stream,
                     rms_w, w_pre, w_post, w_res, wcw);

  size_t shmem = (size_t)2 * BUF_FLOATS * sizeof(float);  // 101376 B (double buffer)
  hipLaunchKernelGGL(mhc_main, dim3(NBLK), dim3(THREADS), shmem, stream,
                     x, wcw, b_pre, b_post, b_res, a_pre, a_post, a_res, out);
}